// GINE_36910948942292
// MI455X (gfx1250) — compile-verified
//
#include <hip/hip_runtime.h>

#define NNODES 50000
#define NEDGES 800000
#define DD 128
#define EDIM 16
#define DEPTH 3
#define PITCH 132   // 128 + 4 pad: row stride % 64 banks = 4 -> conflict-free column reads

typedef float v2f __attribute__((ext_vector_type(2)));
typedef float v8f __attribute__((ext_vector_type(8)));

// ---------------------------------------------------------------- zero aggr
__global__ __launch_bounds__(256) void zero_kernel(float* __restrict__ p, int n) {
    int i = blockIdx.x * blockDim.x + threadIdx.x;
    int stride = gridDim.x * blockDim.x;
    for (; i < n; i += stride) p[i] = 0.0f;
}

// ------------------------------------------------- edge message + scatter-add
// One wave32 per edge. Lane holds 4 of the 128 output features.
__global__ __launch_bounds__(256) void edge_kernel(
    const float* __restrict__ x, const float* __restrict__ edge_attr,
    const long long* __restrict__ eidx, const float* __restrict__ We,
    const float* __restrict__ be, float* __restrict__ aggr, int n_edges)
{
    const int w = (int)((blockIdx.x * (unsigned)blockDim.x + threadIdx.x) >> 5);
    if (w >= n_edges) return;
    const int lane = threadIdx.x & 31;

    const long long src = eidx[w];
    const long long dst = eidx[n_edges + w];

    // lanes 0..15 hold the 16 edge features; broadcast via shuffles
    const float eav = edge_attr[(long long)w * EDIM + (lane & 15)];

    float4 p = *(const float4*)(be + lane * 4);
    #pragma unroll
    for (int k = 0; k < EDIM; ++k) {
        const float a = __shfl(eav, k, 32);
        const float4 wv = *(const float4*)(We + k * DD + lane * 4);
        p.x = fmaf(a, wv.x, p.x);
        p.y = fmaf(a, wv.y, p.y);
        p.z = fmaf(a, wv.z, p.z);
        p.w = fmaf(a, wv.w, p.w);
    }

    const float4 xv = *(const float4*)(x + src * DD + lane * 4);
    float4 m;
    m.x = fmaxf(xv.x + p.x, 0.0f);
    m.y = fmaxf(xv.y + p.y, 0.0f);
    m.z = fmaxf(xv.z + p.z, 0.0f);
    m.w = fmaxf(xv.w + p.w, 0.0f);

    float* ag = aggr + dst * DD + lane * 4;
    unsafeAtomicAdd(ag + 0, m.x);   // native global_atomic_add_f32; aggr is L2-resident
    unsafeAtomicAdd(ag + 1, m.y);
    unsafeAtomicAdd(ag + 2, m.z);
    unsafeAtomicAdd(ag + 3, m.w);
}

// ---------------------------------------------------------------- node update
__device__ __forceinline__ float gelu_exact(float v) {
    return 0.5f * v * (1.0f + erff(v * 0.70710678118654752f));
}

// Fused: h = (1+eps)*x + aggr;  h = gelu(h@W1 + b1);  out = h@W2 + b2
// One wave per 16-row tile, fp32 WMMA 16x16x4, K = 128 (32 steps), N = 128 (8 tiles).
__global__ __launch_bounds__(128) void node_kernel(
    const float* __restrict__ xin, const float* __restrict__ aggr,
    const float* __restrict__ W1, const float* __restrict__ b1,
    const float* __restrict__ W2, const float* __restrict__ b2,
    const float* __restrict__ eps_all, int layer,
    float* __restrict__ xout, int n_nodes)
{
    __shared__ float lds[4][16][PITCH];   // per-wave 16x128 tile (padded)

    const int wave = threadIdx.x >> 5;
    const int lane = threadIdx.x & 31;
    const int lo   = lane & 15;
    const int hi   = lane >> 4;           // 0: K/M 0..1 half, 1: K/M 2..3 half
    const int m0   = (blockIdx.x * 4 + wave) * 16;
    if (m0 >= n_nodes) return;            // wave-uniform; no block-wide sync used

    const float epsv = 1.0f + eps_all[layer];

    // ---- stage h_in = (1+eps)*x + aggr into LDS (coalesced float4 loads)
    #pragma unroll
    for (int r = 0; r < 16; ++r) {
        const long long off = (long long)(m0 + r) * DD + lane * 4;
        const float4 xv = *(const float4*)(xin + off);
        const float4 av = *(const float4*)(aggr + off);
        float4 h;
        h.x = fmaf(epsv, xv.x, av.x);
        h.y = fmaf(epsv, xv.y, av.y);
        h.z = fmaf(epsv, xv.z, av.z);
        h.w = fmaf(epsv, xv.w, av.w);
        *(float4*)&lds[wave][r][lane * 4] = h;
    }
    // same-wave LDS ops are in-order (DScnt): no barrier needed

    // ---- GEMM1: acc = h_in @ W1 + b1
    v8f acc[8];
    #pragma unroll
    for (int nt = 0; nt < 8; ++nt) {
        const float bv = b1[nt * 16 + lo];
        #pragma unroll
        for (int i = 0; i < 8; ++i) acc[nt][i] = bv;
    }
    for (int k0 = 0; k0 < DD; k0 += 4) {
        // A frag: lane(lo,hi) holds A[lo][k0+2*hi], A[lo][k0+2*hi+1]
        const v2f a = *(const v2f*)&lds[wave][lo][k0 + 2 * hi];
        #pragma unroll
        for (int nt = 0; nt < 8; ++nt) {
            v2f b;
            b.x = W1[(k0 + 2 * hi)     * DD + nt * 16 + lo];
            b.y = W1[(k0 + 2 * hi + 1) * DD + nt * 16 + lo];
            acc[nt] = __builtin_amdgcn_wmma_f32_16x16x4_f32(
                false, a, false, b, (short)0, acc[nt], false, false);
        }
    }

    // ---- GELU, write back to LDS transposed into A-layout for GEMM2
    #pragma unroll
    for (int nt = 0; nt < 8; ++nt) {
        #pragma unroll
        for (int r = 0; r < 8; ++r) {
            // C layout: VGPR r -> row r + 8*hi, col nt*16 + lo
            lds[wave][r + 8 * hi][nt * 16 + lo] = gelu_exact(acc[nt][r]);
        }
    }

    // ---- GEMM2: acc2 = h_mid @ W2 + b2
    v8f acc2[8];
    #pragma unroll
    for (int nt = 0; nt < 8; ++nt) {
        const float bv = b2[nt * 16 + lo];
        #pragma unroll
        for (int i = 0; i < 8; ++i) acc2[nt][i] = bv;
    }
    for (int k0 = 0; k0 < DD; k0 += 4) {
        const v2f a = *(const v2f*)&lds[wave][lo][k0 + 2 * hi];
        #pragma unroll
        for (int nt = 0; nt < 8; ++nt) {
            v2f b;
            b.x = W2[(k0 + 2 * hi)     * DD + nt * 16 + lo];
            b.y = W2[(k0 + 2 * hi + 1) * DD + nt * 16 + lo];
            acc2[nt] = __builtin_amdgcn_wmma_f32_16x16x4_f32(
                false, a, false, b, (short)0, acc2[nt], false, false);
        }
    }

    // ---- store output tile
    #pragma unroll
    for (int nt = 0; nt < 8; ++nt) {
        #pragma unroll
        for (int r = 0; r < 8; ++r) {
            xout[(long long)(m0 + r + 8 * hi) * DD + nt * 16 + lo] = acc2[nt][r];
        }
    }
}

// ----------------------------------------------------------------- launcher
extern "C" void kernel_launch(void* const* d_in, const int* in_sizes, int n_in,
                              void* d_out, int out_size, void* d_ws, size_t ws_size,
                              hipStream_t stream) {
    const float* x0        = (const float*)d_in[0];
    const float* edge_attr = (const float*)d_in[1];
    const float* W1        = (const float*)d_in[2];
    const float* b1        = (const float*)d_in[3];
    const float* W2        = (const float*)d_in[4];
    const float* b2        = (const float*)d_in[5];
    const float* We        = (const float*)d_in[6];
    const float* be        = (const float*)d_in[7];
    const float* eps       = (const float*)d_in[8];
    const long long* eidx  = (const long long*)d_in[9];   // int64 in reference
    float* out = (float*)d_out;

    float* aggr = (float*)d_ws;
    float* xA   = aggr + (size_t)NNODES * DD;
    float* xB   = xA   + (size_t)NNODES * DD;

    const int ND = NNODES * DD;
    const int tiles  = (NNODES + 15) / 16;   // 3125
    const int nblocks = (tiles + 3) / 4;     // 4 tiles (waves) per 128-thread block

    for (int l = 0; l < DEPTH; ++l) {
        const float* xin = (l == 0) ? x0 : ((l == 1) ? xA : xB);
        float* xnext     = (l == 0) ? xA : ((l == 1) ? xB : out);

        zero_kernel<<<2048, 256, 0, stream>>>(aggr, ND);
        edge_kernel<<<NEDGES / 8, 256, 0, stream>>>(
            xin, edge_attr, eidx,
            We + (size_t)l * EDIM * DD, be + (size_t)l * DD, aggr, NEDGES);
        node_kernel<<<nblocks, 128, 0, stream>>>(
            xin, aggr,
            W1 + (size_t)l * DD * DD, b1 + (size_t)l * DD,
            W2 + (size_t)l * DD * DD, b2 + (size_t)l * DD,
            eps, l, xnext, NNODES);
    }
}